// MGP_16879221473400
// MI455X (gfx1250) — compile-verified
//
#include <hip/hip_runtime.h>
#include <math.h>

typedef float v2f __attribute__((ext_vector_type(2)));
typedef float v8f __attribute__((ext_vector_type(8)));

#define DXD 8
#define PD  10
#define MTR 256
#define TP  16
#define LAMP 10000.0f
#define JIT  1e-6f

// ---------------- workspace layout (float offsets) ----------------
enum : int {
  OFF_XT    = 0,                         // [256][8]  X / lengthscale
  OFF_KT    = OFF_XT + MTR*DXD,          // [256][256] noiseless RBF Ktil
  OFF_L     = OFF_KT + MTR*MTR,          // Kmm -> chol(L) in-place (lower)
  OFF_LINV  = OFF_L + MTR*MTR,           // L^-1 (lower)
  OFF_KINV  = OFF_LINV + MTR*MTR,        // K^-1
  OFF_ALPHA = OFF_KINV + MTR*MTR,        // [256]
  OFF_BETA  = OFF_ALPHA + MTR,           // [m*10+p]  beta_p = Kinv G_p alpha
  OFF_GV    = OFF_BETA + MTR*PD,         // [p*256+m] g_p = G_p alpha
  OFF_W     = OFF_GV + MTR*PD,           // [256][256] tmp
  OFF_MTMP  = OFF_W + MTR*MTR,           // [256][256] tmp
  OFF_HC    = OFF_MTMP + MTR*MTR,        // [100] tr(Kinv G_p Kinv G_q)
  OFF_HBD   = OFF_HC + 100,              // [92] quadratic-form/trace partials
  OFF_LH    = OFF_HBD + 100,             // [100] chol(-hess) lower 10x10
  WS_FLOATS = OFF_LH + 100
};

#define TRI(p,q) ((p)*8 + (q) - ((p)*((p)+1))/2)   // p<=q<8 -> 0..35

// ---------------- wave helpers (wave32) ----------------
__device__ inline float r16(float v) {          // reduce within 16-lane groups
  v += __shfl_xor(v, 1, 32);
  v += __shfl_xor(v, 2, 32);
  v += __shfl_xor(v, 4, 32);
  v += __shfl_xor(v, 8, 32);
  return v;
}
__device__ inline float rwave(float v) {        // reduce across full wave32
  v = r16(v);
  v += __shfl_xor(v, 16, 32);
  return v;
}

// ---------------- WMMA GEMM: OUT[16 t][256 m] = Kmat[256x256] @ IN[16 t][256 m]
// 256-thread block = 8 waves; wave w computes M-tiles {2w, 2w+1}.
// A frag: M = lane%16, K = 2*(lane/16)+g ; B frag: N = lane%16, K = 2*(lane/16)+g
// D frag: M = g + 8*(lane/16), N = lane%16
__device__ inline void wgemm(const float* __restrict__ Kmat,
                             const float* sIn, float* sOut) {
  const int tid  = threadIdx.x;
  const int lane = tid & 31;
  const int wave = tid >> 5;
  const int hf   = lane >> 4;
  const int l16  = lane & 15;
#pragma unroll
  for (int rep = 0; rep < 2; ++rep) {
    const int mt = wave * 2 + rep;
    const float* Arow = Kmat + (mt * 16 + l16) * MTR + hf * 2;
    const float* Brow = sIn + l16 * MTR + hf * 2;
    v8f acc;
#pragma unroll
    for (int g = 0; g < 8; ++g) acc[g] = 0.0f;
#pragma unroll 4
    for (int kk = 0; kk < MTR; kk += 4) {
      v2f a = *(const v2f*)(Arow + kk);
      v2f b = *(const v2f*)(Brow + kk);
      acc = __builtin_amdgcn_wmma_f32_16x16x4_f32(false, a, false, b,
                                                  (short)0, acc, false, false);
    }
    float* o = sOut + l16 * MTR + mt * 16 + hf * 8;
#pragma unroll
    for (int g = 0; g < 8; ++g) o[g] = acc[g];
  }
}

// ================= setup kernels =================

// XT[m][d] = X[m][d] * exp(-theta[d])
__global__ __launch_bounds__(256) void k_prep(const float* __restrict__ X,
                                              const float* __restrict__ th,
                                              float* __restrict__ ws) {
  int m = threadIdx.x;
  for (int d = 0; d < DXD; ++d)
    ws[OFF_XT + m * DXD + d] = X[m * DXD + d] * expf(-th[d]);
}

// KT = var*exp(-0.5 d2) ; L-buffer = KT + (noise+jitter) I
__global__ __launch_bounds__(256) void k_ker(const float* __restrict__ th,
                                             float* __restrict__ ws) {
  int m = blockIdx.x, n = threadIdx.x;
  const float* XT = ws + OFF_XT;
  float var = expf(th[DXD]), noise = expf(th[DXD + 1]);
  float d2 = 0.f;
#pragma unroll
  for (int d = 0; d < DXD; ++d) {
    float df = XT[m * DXD + d] - XT[n * DXD + d];
    d2 += df * df;
  }
  float kt = var * expf(-0.5f * d2);
  ws[OFF_KT + m * MTR + n] = kt;
  ws[OFF_L + m * MTR + n] = kt + ((m == n) ? (noise + JIT) : 0.f);
}

// in-place 256x256 Cholesky (lower) on OFF_L, one block
__global__ __launch_bounds__(256) void k_chol(float* __restrict__ ws) {
  float* A = ws + OFF_L;
  int i = threadIdx.x;
  for (int k = 0; k < MTR; ++k) {
    if (i == k) A[k * MTR + k] = sqrtf(A[k * MTR + k]);
    __syncthreads();
    if (i > k) A[i * MTR + k] /= A[k * MTR + k];
    __syncthreads();
    if (i > k) {
      float lik = A[i * MTR + k];
      for (int j = k + 1; j <= i; ++j) A[i * MTR + j] -= lik * A[j * MTR + k];
    }
    __syncthreads();
  }
}

// LINV: column-parallel forward substitution of L Z = I
__global__ __launch_bounds__(256) void k_linv(float* __restrict__ ws) {
  const float* L = ws + OFF_L;
  float* Z = ws + OFF_LINV;
  int c = threadIdx.x;
  for (int i = 0; i < c; ++i) Z[i * MTR + c] = 0.f;
  for (int i = c; i < MTR; ++i) {
    float s = (i == c) ? 1.f : 0.f;
    for (int k = c; k < i; ++k) s -= L[i * MTR + k] * Z[k * MTR + c];
    Z[i * MTR + c] = s / L[i * MTR + i];
  }
}

// KINV = LINV^T LINV
__global__ __launch_bounds__(256) void k_kinv(float* __restrict__ ws) {
  int i = blockIdx.x, j = threadIdx.x;
  const float* Z = ws + OFF_LINV;
  int m0 = (i > j) ? i : j;
  float s = 0.f;
  for (int m = m0; m < MTR; ++m) s += Z[m * MTR + i] * Z[m * MTR + j];
  ws[OFF_KINV + i * MTR + j] = s;
}

// alpha = KINV @ Y
__global__ __launch_bounds__(256) void k_alpha(const float* __restrict__ Y,
                                               float* __restrict__ ws) {
  int m = threadIdx.x;
  const float* KI = ws + OFF_KINV;
  float s = 0.f;
  for (int n = 0; n < MTR; ++n) s += KI[m * MTR + n] * Y[n];
  ws[OFF_ALPHA + m] = s;
}

// g_p = G_p alpha  (G_p = dK/dtheta_p)
__global__ __launch_bounds__(256) void k_gv(const float* __restrict__ th,
                                            float* __restrict__ ws) {
  int m = threadIdx.x;
  const float* KT = ws + OFF_KT;
  const float* XT = ws + OFF_XT;
  const float* AL = ws + OFF_ALPHA;
  float noise = expf(th[DXD + 1]);
  float g[PD];
#pragma unroll
  for (int p = 0; p < PD; ++p) g[p] = 0.f;
  float xm[DXD];
#pragma unroll
  for (int d = 0; d < DXD; ++d) xm[d] = XT[m * DXD + d];
  for (int n = 0; n < MTR; ++n) {
    float ka = KT[m * MTR + n] * AL[n];
#pragma unroll
    for (int d = 0; d < DXD; ++d) {
      float df = xm[d] - XT[n * DXD + d];
      g[d] += ka * df * df;
    }
    g[8] += ka;
  }
  g[9] = noise * AL[m];
#pragma unroll
  for (int p = 0; p < PD; ++p) ws[OFF_GV + p * MTR + m] = g[p];
}

// beta_p = KINV g_p  (stored [m*10+p])
__global__ __launch_bounds__(256) void k_beta(float* __restrict__ ws) {
  int m = threadIdx.x;
  const float* KI = ws + OFF_KINV;
  const float* GV = ws + OFF_GV;
  float b[PD];
#pragma unroll
  for (int p = 0; p < PD; ++p) b[p] = 0.f;
  for (int n = 0; n < MTR; ++n) {
    float kin = KI[m * MTR + n];
#pragma unroll
    for (int p = 0; p < PD; ++p) b[p] += kin * GV[p * MTR + n];
  }
#pragma unroll
  for (int p = 0; p < PD; ++p) ws[OFF_BETA + m * PD + p] = b[p];
}

// Hessian quadratic-form / trace partials (terms B and D pieces)
__global__ __launch_bounds__(256) void k_hbd(float* __restrict__ ws) {
  int tid = threadIdx.x, m = tid;
  const float* KT = ws + OFF_KT;
  const float* KI = ws + OFF_KINV;
  const float* XT = ws + OFF_XT;
  const float* AL = ws + OFF_ALPHA;
  float vals[92];
#pragma unroll
  for (int i = 0; i < 92; ++i) vals[i] = 0.f;
  float am = AL[m];
  vals[90] = KI[m * MTR + m];  // -> tr(Kinv)
  vals[91] = am * am;          // -> sum alpha^2
  float xm[DXD];
#pragma unroll
  for (int d = 0; d < DXD; ++d) xm[d] = XT[m * DXD + d];
  for (int n = 0; n < MTR; ++n) {
    float kt = KT[m * MTR + n];
    float waa = am * AL[n] * kt;
    float wkd = KI[m * MTR + n] * kt;
    float s[DXD];
#pragma unroll
    for (int d = 0; d < DXD; ++d) {
      float df = xm[d] - XT[n * DXD + d];
      s[d] = df * df;
    }
#pragma unroll
    for (int p = 0; p < 8; ++p) {
      vals[72 + p] += waa * s[p];
      vals[80 + p] += wkd * s[p];
#pragma unroll
      for (int q = p; q < 8; ++q) {
        float sq = s[p] * s[q];
        vals[TRI(p, q)] += waa * sq;
        vals[36 + TRI(p, q)] += wkd * sq;
      }
    }
    vals[88] += waa;
    vals[89] += wkd;
  }
  __shared__ float part[8][92];
  int wave = tid >> 5, lane = tid & 31;
#pragma unroll
  for (int i = 0; i < 92; ++i) vals[i] = rwave(vals[i]);
  if (lane == 0)
#pragma unroll
    for (int i = 0; i < 92; ++i) part[wave][i] = vals[i];
  __syncthreads();
  if (tid < 92) {
    float s = 0.f;
    for (int w = 0; w < 8; ++w) s += part[w][tid];
    ws[OFF_HBD + tid] = s;
  }
}

// W = KINV @ G_p
__global__ __launch_bounds__(256) void k_wg(const float* __restrict__ th,
                                            float* __restrict__ ws, int p) {
  int i = blockIdx.x, j = threadIdx.x;
  const float* KI = ws + OFF_KINV;
  const float* KT = ws + OFF_KT;
  const float* XT = ws + OFF_XT;
  float noise = expf(th[DXD + 1]);
  float s = 0.f;
  if (p == 9) {
    s = noise * KI[i * MTR + j];
  } else if (p == 8) {
    for (int k = 0; k < MTR; ++k) s += KI[i * MTR + k] * KT[k * MTR + j];
  } else {
    float xj = XT[j * DXD + p];
    for (int k = 0; k < MTR; ++k) {
      float df = XT[k * DXD + p] - xj;
      s += KI[i * MTR + k] * KT[k * MTR + j] * df * df;
    }
  }
  ws[OFF_W + i * MTR + j] = s;
}

// MTMP = W @ KINV
__global__ __launch_bounds__(256) void k_mg(float* __restrict__ ws) {
  int i = blockIdx.x, j = threadIdx.x;
  const float* W = ws + OFF_W;
  const float* KI = ws + OFF_KINV;
  float s = 0.f;
  for (int k = 0; k < MTR; ++k) s += W[i * MTR + k] * KI[k * MTR + j];
  ws[OFF_MTMP + i * MTR + j] = s;
}

// HC[p][q] = <MTMP, G_q>_F = tr(Kinv G_p Kinv G_q)
__global__ __launch_bounds__(256) void k_hc(const float* __restrict__ th,
                                            float* __restrict__ ws, int p) {
  int tid = threadIdx.x, m = tid;
  const float* MT_ = ws + OFF_MTMP;
  const float* KT = ws + OFF_KT;
  const float* XT = ws + OFF_XT;
  float noise = expf(th[DXD + 1]);
  float acc[PD];
#pragma unroll
  for (int q = 0; q < PD; ++q) acc[q] = 0.f;
  float xm[DXD];
#pragma unroll
  for (int d = 0; d < DXD; ++d) xm[d] = XT[m * DXD + d];
  for (int n = 0; n < MTR; ++n) {
    float mk = MT_[m * MTR + n] * KT[m * MTR + n];
#pragma unroll
    for (int q = 0; q < 8; ++q) {
      float df = xm[q] - XT[n * DXD + q];
      acc[q] += mk * df * df;
    }
    acc[8] += mk;
  }
  acc[9] = MT_[m * MTR + m] * noise;
  __shared__ float part[8][PD];
  int wave = tid >> 5, lane = tid & 31;
#pragma unroll
  for (int q = 0; q < PD; ++q) acc[q] = rwave(acc[q]);
  if (lane == 0)
#pragma unroll
    for (int q = 0; q < PD; ++q) part[wave][q] = acc[q];
  __syncthreads();
  if (tid < PD) {
    float s = 0.f;
    for (int w = 0; w < 8; ++w) s += part[w][tid];
    ws[OFF_HC + p * PD + tid] = s;
  }
}

// assemble -hess(obj) and 10x10 Cholesky -> LH
__global__ __launch_bounds__(128) void k_hess(const float* __restrict__ th,
                                              float* __restrict__ ws) {
  __shared__ float ha[100];
  int tid = threadIdx.x;
  const float* GV = ws + OFF_GV;
  const float* BE = ws + OFF_BETA;
  if (tid < 100) {
    int p = tid / PD, q = tid % PD;
    float s = 0.f;
    for (int m = 0; m < MTR; ++m) s += GV[p * MTR + m] * BE[m * PD + q];
    ha[tid] = s;  // g_p^T Kinv g_q
  }
  __syncthreads();
  if (tid == 0) {
    const float* HB = ws + OFF_HBD;
    const float* HC = ws + OFF_HC;
    float noise = expf(th[DXD + 1]);
    float A[100];
    for (int p = 0; p < PD; ++p)
      for (int q = 0; q < PD; ++q) {
        int a = (p < q) ? p : q, b = (p < q) ? q : p;
        float B = 0.f, D = 0.f;
        if (b < 8) {
          int id = TRI(a, b);
          B = 0.5f * HB[id];
          D = 0.5f * HB[36 + id];
          if (a == b) { B -= HB[72 + a]; D -= HB[80 + a]; }
        } else if (b == 8) {
          if (a < 8)      { B = 0.5f * HB[72 + a]; D = 0.5f * HB[80 + a]; }
          else            { B = 0.5f * HB[88];     D = 0.5f * HB[89]; }
        } else {  // b == 9
          if (a == 9)     { B = 0.5f * noise * HB[91]; D = 0.5f * noise * HB[90]; }
        }
        float haSym = 0.5f * (ha[p * PD + q] + ha[q * PD + p]);
        float hcSym = 0.25f * (HC[p * PD + q] + HC[q * PD + p]);
        float hval = -haSym + B + 2.0f * hcSym * 0.5f * 0.5f + 0.5f *
                     (HC[p * PD + q] * 0.f) - D;  // keep simple below
        hval = -haSym + B + 0.5f * (hcSym * 2.0f) - D;  // 0.5*tr(C), symmetrized
        A[p * PD + q] = ((p == q) ? LAMP : 0.f) - hval;  // -hess(obj)
      }
    // 10x10 Cholesky (lower)
    for (int k = 0; k < PD; ++k) {
      A[k * PD + k] = sqrtf(A[k * PD + k]);
      for (int i = k + 1; i < PD; ++i) A[i * PD + k] /= A[k * PD + k];
      for (int i = k + 1; i < PD; ++i)
        for (int j = k + 1; j <= i; ++j) A[i * PD + j] -= A[i * PD + k] * A[j * PD + k];
    }
    for (int i = 0; i < 100; ++i) ws[OFF_LH + i] = A[i];
  }
}

// ================= main heavy kernel (WMMA) =================
__global__ __launch_bounds__(256) void k_main(const float* __restrict__ Xnew,
                                              const float* __restrict__ th,
                                              const float* __restrict__ ws,
                                              float* __restrict__ out, int N) {
  __shared__ float sA[TP * MTR];   // k, later u_p
  __shared__ float sB[TP * MTR];   // c = K^-1 k
  __shared__ float sC[TP * MTR];   // r = Ktil c, later q_p
  __shared__ float sXn[TP * DXD];
  __shared__ float red[TP][64];

  const float* XT = ws + OFF_XT;
  const float* KT = ws + OFF_KT;
  const float* KI = ws + OFF_KINV;
  const float* AL = ws + OFF_ALPHA;
  const float* BE = ws + OFF_BETA;
  const float* LH = ws + OFF_LH;

  const int tid = threadIdx.x;
  const int base = blockIdx.x * TP;
  const float var = expf(th[DXD]);
  const float noise = expf(th[DXD + 1]);

  if (tid < TP * DXD) {
    int t = tid >> 3, d = tid & 7;
    sXn[tid] = Xnew[(base + t) * DXD + d] * expf(-th[d]);
  }
  __syncthreads();

  // ---- k columns: sA[t][m] ----
  {
    float xm[DXD];
#pragma unroll
    for (int d = 0; d < DXD; ++d) xm[d] = XT[tid * DXD + d];
    for (int t = 0; t < TP; ++t) {
      float d2 = 0.f;
#pragma unroll
      for (int d = 0; d < DXD; ++d) {
        float df = xm[d] - sXn[t * DXD + d];
        d2 += df * df;
      }
      sA[t * MTR + tid] = var * expf(-0.5f * d2);
    }
  }
  __syncthreads();

  wgemm(KI, sA, sB);   // c
  __syncthreads();
  wgemm(KT, sB, sC);   // r
  __syncthreads();

  // ---- per-test-point moment reductions (16 threads per point) ----
  const int t = tid >> 4, l = tid & 15;
  {
    float fm = 0, kc = 0, cc = 0, cr = 0, S0 = 0;
    float S1[8] = {0}, S2[8] = {0}, T1[8] = {0}, T2[8] = {0};
    float kb[PD] = {0}, cx[8] = {0};
    for (int j = 0; j < 16; ++j) {
      int m = l + j * 16;
      float kv = sA[t * MTR + m], cv = sB[t * MTR + m], rv = sC[t * MTR + m];
      float a = AL[m];
      float w = kv * a, wp = kv * cv;
      fm += w; kc += wp; cc += cv * cv; cr += cv * rv; S0 += w;
#pragma unroll
      for (int d = 0; d < 8; ++d) {
        float x = XT[m * DXD + d];
        float wx = w * x, wpx = wp * x;
        S1[d] += wx; S2[d] += wx * x;
        T1[d] += wpx; T2[d] += wpx * x;
        cx[d] += cv * x * x * rv;
      }
#pragma unroll
      for (int p = 0; p < PD; ++p) kb[p] += kv * BE[m * PD + p];
    }
    fm = r16(fm); kc = r16(kc); cc = r16(cc); cr = r16(cr); S0 = r16(S0);
#pragma unroll
    for (int d = 0; d < 8; ++d) {
      S1[d] = r16(S1[d]); S2[d] = r16(S2[d]);
      T1[d] = r16(T1[d]); T2[d] = r16(T2[d]); cx[d] = r16(cx[d]);
    }
#pragma unroll
    for (int p = 0; p < PD; ++p) kb[p] = r16(kb[p]);
    if (l == 0) {
      red[t][0] = fm; red[t][1] = kc; red[t][2] = cc; red[t][3] = cr; red[t][4] = S0;
#pragma unroll
      for (int d = 0; d < 8; ++d) {
        red[t][6 + d] = S1[d]; red[t][14 + d] = S2[d];
        red[t][22 + d] = T1[d]; red[t][30 + d] = T2[d]; red[t][48 + d] = cx[d];
      }
#pragma unroll
      for (int p = 0; p < PD; ++p) red[t][38 + p] = kb[p];
    }
  }

  // ---- u_p^T Ktil u_p per lengthscale ----
  for (int p = 0; p < 8; ++p) {
    __syncthreads();
    {
      float xp = XT[tid * DXD + p];
      for (int t2 = 0; t2 < TP; ++t2)
        sA[t2 * MTR + tid] = sB[t2 * MTR + tid] * xp;
    }
    __syncthreads();
    wgemm(KT, sA, sC);   // q_p
    __syncthreads();
    float s = 0.f;
    for (int j = 0; j < 16; ++j) {
      int m = l + j * 16;
      s += sA[t * MTR + m] * sC[t * MTR + m];
    }
    s = r16(s);
    if (l == 0) red[t][56 + p] = s;
  }
  __syncthreads();

  // ---- final assembly: gradients, 10x10 triangular solves, mvar ----
  if (tid < TP) {
    int tt = tid;
    float S0 = red[tt][4], kc = red[tt][1];
    float g1[PD], g2[PD];
#pragma unroll
    for (int p = 0; p < 8; ++p) {
      float x = sXn[tt * DXD + p];
      g1[p] = (red[tt][14 + p] - 2.f * x * red[tt][6 + p] + x * x * S0)
              - red[tt][38 + p];
      float dkc = red[tt][30 + p] - 2.f * x * red[tt][22 + p] + x * x * kc;
      g2[p] = -2.f * dkc + 2.f * (red[tt][48 + p] - red[tt][56 + p]);
    }
    g1[8] = S0 - red[tt][46];
    g1[9] = -red[tt][47];
    g2[8] = var - 2.f * kc + red[tt][3];
    g2[9] = noise * red[tt][2];
    float fvar = var - kc;

    float z1[PD], z2[PD], s1 = 0.f, s2 = 0.f;
    for (int i = 0; i < PD; ++i) {
      float a1 = g1[i], a2 = g2[i];
      for (int j = 0; j < i; ++j) {
        float lij = LH[i * PD + j];
        a1 -= lij * z1[j];
        a2 -= lij * z2[j];
      }
      float dinv = 1.f / LH[i * PD + i];
      z1[i] = a1 * dinv; z2[i] = a2 * dinv;
      s1 += z1[i] * z1[i]; s2 += z2[i] * z2[i];
    }
    float mvar = (4.f / 3.f) * fvar + s1 + (1.f / 3.f) / (fvar + 0.001f) * s2;
    out[base + tt] = red[tt][0];
    out[N + base + tt] = mvar;
  }
}

// ================= launcher =================
extern "C" void kernel_launch(void* const* d_in, const int* in_sizes, int n_in,
                              void* d_out, int out_size, void* d_ws, size_t ws_size,
                              hipStream_t stream) {
  const float* X    = (const float*)d_in[0];
  const float* Y    = (const float*)d_in[1];
  const float* Xnew = (const float*)d_in[2];
  const float* th   = (const float*)d_in[3];
  float* ws = (float*)d_ws;
  float* out = (float*)d_out;
  int N = in_sizes[2] / DXD;

  k_prep<<<1, 256, 0, stream>>>(X, th, ws);
  k_ker<<<MTR, 256, 0, stream>>>(th, ws);
  k_chol<<<1, 256, 0, stream>>>(ws);
  k_linv<<<1, 256, 0, stream>>>(ws);
  k_kinv<<<MTR, 256, 0, stream>>>(ws);
  k_alpha<<<1, 256, 0, stream>>>(Y, ws);
  k_gv<<<1, 256, 0, stream>>>(th, ws);
  k_beta<<<1, 256, 0, stream>>>(ws);
  k_hbd<<<1, 256, 0, stream>>>(ws);
  for (int p = 0; p < PD; ++p) {
    k_wg<<<MTR, 256, 0, stream>>>(th, ws, p);
    k_mg<<<MTR, 256, 0, stream>>>(ws);
    k_hc<<<1, 256, 0, stream>>>(th, ws, p);
  }
  k_hess<<<1, 128, 0, stream>>>(th, ws);
  k_main<<<N / TP, 256, 0, stream>>>(Xnew, th, ws, out, N);
}